// InstanceSegmentationLoss_15049565405346
// MI455X (gfx1250) — compile-verified
//
#include <hip/hip_runtime.h>

#define NUM_LABELS 65
#define NBINS (NUM_LABELS * NUM_LABELS)   // 4225
#define PADL 80                           // 5 * 16, WMMA-friendly padding
#define HW (2048 * 2048)

typedef __attribute__((ext_vector_type(2))) float v2f;
typedef __attribute__((ext_vector_type(8))) float v8f;

// ---------------------------------------------------------------------------
// Kernel 0: zero the global confusion matrix (d_ws is poisoned by harness)
// ---------------------------------------------------------------------------
__global__ void zero_conf_kernel(unsigned* __restrict__ conf) {
  int i = blockIdx.x * blockDim.x + threadIdx.x;
  if (i < NBINS) conf[i] = 0u;
}

// ---------------------------------------------------------------------------
// Kernel 1: streaming co-occurrence histogram.
// 32 MiB of int32 reads -> memory bound (~1.4 us at 23.3 TB/s).
// LDS-privatized 4225-bin histogram (ds_add_u32), integer flush atomics
// (exact, commutative -> deterministic regardless of schedule).
// ---------------------------------------------------------------------------
__global__ __launch_bounds__(256) void hist_kernel(const int* __restrict__ pred,
                                                   const int* __restrict__ truem,
                                                   unsigned* __restrict__ conf,
                                                   int n4) {
  __shared__ unsigned lconf[NBINS];
  for (int i = threadIdx.x; i < NBINS; i += blockDim.x) lconf[i] = 0u;
  __syncthreads();

  const int4* p4 = (const int4*)pred;   // global_load_b128 stream
  const int4* t4 = (const int4*)truem;
  int stride = gridDim.x * blockDim.x;
  for (int i = blockIdx.x * blockDim.x + threadIdx.x; i < n4; i += stride) {
    int4 p = p4[i];
    int4 t = t4[i];
    atomicAdd(&lconf[p.x * NUM_LABELS + t.x], 1u);
    atomicAdd(&lconf[p.y * NUM_LABELS + t.y], 1u);
    atomicAdd(&lconf[p.z * NUM_LABELS + t.z], 1u);
    atomicAdd(&lconf[p.w * NUM_LABELS + t.w], 1u);
  }
  __syncthreads();

  for (int i = threadIdx.x; i < NBINS; i += blockDim.x) {
    unsigned v = lconf[i];
    if (v) atomicAdd(&conf[i], v);
  }
}

// ---------------------------------------------------------------------------
// Kernel 2: finisher (single workgroup).
// area_pred = conf * ones, area_true = conf^T * ones via V_WMMA_F32_16X16X4_F32
// on an 80x80 zero-padded f32 copy of conf in LDS (counts <= 2^22 -> exact).
// Then 128 threads compute best-IoU per row/col, thread 0 reduces.
// ---------------------------------------------------------------------------
__global__ __launch_bounds__(256) void finish_kernel(const unsigned* __restrict__ conf,
                                                     float* __restrict__ out) {
  __shared__ float cf[PADL * PADL];
  __shared__ float areaP[PADL];
  __shared__ float areaT[PADL];
  __shared__ float contrib[128];
  int tid = threadIdx.x;

  for (int i = tid; i < PADL * PADL; i += 256) {
    int r = i / PADL, c = i % PADL;
    cf[i] = (r < NUM_LABELS && c < NUM_LABELS) ? (float)conf[r * NUM_LABELS + c] : 0.0f;
  }
  __syncthreads();

  // Wave 0 only (tid<32 is wave-uniform under wave32) -> EXEC all-1s for WMMA.
  if (tid < 32) {
    int lane = tid;
    int mrow = lane & 15;
    int koff = (lane < 16) ? 0 : 2;   // A 16x4 f32 layout: VGPR0=K{0|2}, VGPR1=K{1|3}
    v2f ones; ones.x = 1.0f; ones.y = 1.0f;  // B = all-ones (layout-invariant)
    for (int band = 0; band < 5; ++band) {
      int row = band * 16 + mrow;
      v8f accP = {};
      v8f accT = {};
      for (int k0 = 0; k0 < PADL; k0 += 4) {
        v2f a, at;
        a.x  = cf[row * PADL + (k0 + koff)];          // A[m,k]   = conf[m,k]
        a.y  = cf[row * PADL + (k0 + koff + 1)];
        at.x = cf[(k0 + koff) * PADL + row];          // A'[m,k]  = conf[k,m]  (transpose)
        at.y = cf[(k0 + koff + 1) * PADL + row];
        accP = __builtin_amdgcn_wmma_f32_16x16x4_f32(
            false, a,  false, ones, (short)0, accP, false, false);
        accT = __builtin_amdgcn_wmma_f32_16x16x4_f32(
            false, at, false, ones, (short)0, accT, false, false);
      }
      // D layout: VGPR r -> M=r (lanes 0-15), M=8+r (lanes 16-31); all N equal.
      if (lane == 0) {
#pragma unroll
        for (int r = 0; r < 8; ++r) {
          areaP[band * 16 + r] = accP[r];
          areaT[band * 16 + r] = accT[r];
        }
      }
      if (lane == 16) {
#pragma unroll
        for (int r = 0; r < 8; ++r) {
          areaP[band * 16 + 8 + r] = accP[r];
          areaT[band * 16 + 8 + r] = accT[r];
        }
      }
    }
  }
  __syncthreads();

  if (tid < 128) {
    bool isRow = tid < 64;            // rows: best_for_pred; cols: best_for_true
    int idx = (tid & 63) + 1;         // label 1..64
    float area = isRow ? areaP[idx] : areaT[idx];
    if (area > 0.0f) {
      float best = 0.0f;
      for (int j = 1; j < NUM_LABELS; ++j) {
        float inter = isRow ? cf[idx * PADL + j] : cf[j * PADL + idx];
        float uni = (isRow ? areaP[idx] + areaT[j] : areaP[j] + areaT[idx]) - inter;
        float iou = (uni > 0.0f) ? inter / fmaxf(uni, 1.0f) : 0.0f;
        best = fmaxf(best, iou);
      }
      contrib[tid] = 1.0f - best;     // present instance contribution (>= 0)
    } else {
      contrib[tid] = -1.0f;           // sentinel: instance absent
    }
  }
  __syncthreads();

  if (tid == 0) {
    float total = 0.0f;
    int n = 0;
    for (int i = 0; i < 128; ++i) {   // fixed order -> deterministic
      float c = contrib[i];
      if (c >= 0.0f) { total += c; n++; }
    }
    out[0] = (n > 0) ? (total / (float)n) : 0.0f;
  }
}

// ---------------------------------------------------------------------------
extern "C" void kernel_launch(void* const* d_in, const int* in_sizes, int n_in,
                              void* d_out, int out_size, void* d_ws, size_t ws_size,
                              hipStream_t stream) {
  const int* pred  = (const int*)d_in[0];
  const int* truem = (const int*)d_in[1];
  unsigned* conf = (unsigned*)d_ws;       // 4225 * 4 B scratch
  float* out = (float*)d_out;
  int n4 = HW / 4;

  zero_conf_kernel<<<(NBINS + 255) / 256, 256, 0, stream>>>(conf);
  hist_kernel<<<512, 256, 0, stream>>>(pred, truem, conf, n4);
  finish_kernel<<<1, 256, 0, stream>>>(conf, out);
}